// FFPS_Sampler_5892695130400
// MI455X (gfx1250) — compile-verified
//
#include <hip/hip_runtime.h>
#include <hip/hip_bf16.h>

typedef __attribute__((ext_vector_type(2))) float v2f;
typedef __attribute__((ext_vector_type(8))) float v8f;

#define B_   4
#define N_   4096
#define C_   128
#define DP_  132          // 3 + 128 padded to multiple of 4 for K-loop
#define TPB_SCAN 1024
#define COLS_ (N_ / TPB_SCAN)   // 4 columns per thread

// ---------------------------------------------------------------------------
// Phase 1: build padded feature matrix (row-major + col-major) and sq norms.
// Frm: [B][N][DP_]   (for WMMA fragment loads, 8B-aligned float2 access)
// Fcm: [B][DP_][N]   (for coalesced GEMV fallback)
// sq : [B][N]
// ---------------------------------------------------------------------------
__global__ void ffps_build(const float* __restrict__ pts,
                           const float* __restrict__ feat,
                           float* __restrict__ Frm,
                           float* __restrict__ Fcm,
                           float* __restrict__ sq) {
  int p = blockIdx.x * blockDim.x + threadIdx.x;
  if (p >= B_ * N_) return;
  int b = p >> 12;
  int n = p & (N_ - 1);
  float acc = 0.f;
  float* frow = Frm + (size_t)p * DP_;
  float* fcb  = Fcm + (size_t)b * DP_ * N_;
#pragma unroll
  for (int d = 0; d < 3; ++d) {
    float v = pts[(size_t)p * 3 + d];
    frow[d] = v;
    fcb[(size_t)d * N_ + n] = v;
    acc += v * v;
  }
  for (int c = 0; c < C_; ++c) {
    float v = feat[((size_t)b * C_ + c) * N_ + n];   // coalesced across n
    int d = 3 + c;
    frow[d] = v;
    fcb[(size_t)d * N_ + n] = v;
    acc += v * v;
  }
  frow[DP_ - 1] = 0.f;
  fcb[(size_t)(DP_ - 1) * N_ + n] = 0.f;
  sq[p] = acc;
}

// ---------------------------------------------------------------------------
// Phase 2: dist[b,i,j] = sq[i] + sq[j] - 2 * <F_i, F_j> via V_WMMA_F32_16X16X4_F32
// One wave32 per 16x16 output tile, 4 waves per block.
// A (16x4 f32): lanes 0-15 hold M=0..15 K={k,k+1}; lanes 16-31 K={k+2,k+3}.
// B (4x16 f32): symmetric Gram -> same load pattern with the j-tile base.
// C/D: VGPR r -> (M = r + 8*lane[4], N = lane&15).
// ---------------------------------------------------------------------------
__global__ void ffps_gram_wmma(const float* __restrict__ Frm,
                               const float* __restrict__ sq,
                               float* __restrict__ dist) {
  const int tilesPerRow   = N_ / 16;                 // 256
  const int tilesPerBatch = tilesPerRow * tilesPerRow;
  int wave = threadIdx.x >> 5;
  int lane = threadIdx.x & 31;
  long gtile = (long)blockIdx.x * 4 + wave;
  int b  = (int)(gtile / tilesPerBatch);
  int t  = (int)(gtile % tilesPerBatch);
  int ti = (t / tilesPerRow) * 16;
  int tj = (t % tilesPerRow) * 16;

  const float* Fb = Frm + (size_t)b * N_ * DP_;
  int lrow  = lane & 15;
  int khalf = (lane >> 4) * 2;
  const float* aptr = Fb + (size_t)(ti + lrow) * DP_ + khalf;
  const float* bptr = Fb + (size_t)(tj + lrow) * DP_ + khalf;

  v8f acc = {};
#pragma unroll 4
  for (int k = 0; k < DP_; k += 4) {
    v2f av = *(const v2f*)(aptr + k);
    v2f bv = *(const v2f*)(bptr + k);
    acc = __builtin_amdgcn_wmma_f32_16x16x4_f32(
        /*neg_a=*/false, av, /*neg_b=*/false, bv,
        /*c_mod=*/(short)0, acc, /*reuse_a=*/false, /*reuse_b=*/false);
  }

  const float* sqb = sq + (size_t)b * N_;
  int col = tj + lrow;
  float sqj = sqb[col];
  int rbase = ti + ((lane >> 4) << 3);
  float* dbase = dist + ((size_t)b * N_ + rbase) * N_ + col;
#pragma unroll
  for (int r = 0; r < 8; ++r) {
    float v = sqb[rbase + r] + sqj - 2.0f * acc[r];
    dbase[(size_t)r * N_] = v;
  }
}

// ---------------------------------------------------------------------------
// Argmax reduce helper (first-occurrence tie-break => prefer smaller index)
// ---------------------------------------------------------------------------
__device__ __forceinline__ void amax2(float& v, int& j, float ov, int oj) {
  if (ov > v || (ov == v && oj < j)) { v = ov; j = oj; }
}
__device__ __forceinline__ void wave_amax(float& v, int& j) {
#pragma unroll
  for (int off = 16; off > 0; off >>= 1) {
    float ov = __shfl_down(v, off, 32);
    int   oj = __shfl_down(j, off, 32);
    amax2(v, j, ov, oj);
  }
}

// ---------------------------------------------------------------------------
// Phase 3a: sequential FPS scan over precomputed dist rows.
// One block per batch, 1024 threads (32 waves), 4 columns/thread in regs.
// ---------------------------------------------------------------------------
__global__ void ffps_scan_pre(const float* __restrict__ dist,
                              int* __restrict__ out, int npoint) {
  int b = blockIdx.x, tid = threadIdx.x;
  int lane = tid & 31, wave = tid >> 5;
  __shared__ float s_wv[32];
  __shared__ int   s_wi[32];
  __shared__ int   s_last;

  float mind[COLS_];
#pragma unroll
  for (int q = 0; q < COLS_; ++q) mind[q] = 1e10f;
  int last = 0;
  const float* db = dist + (size_t)b * N_ * N_;

  for (int s = 0; s < npoint; ++s) {
    if (tid == 0) out[b * npoint + s] = last;
    const float* row = db + (size_t)last * N_;
    float bv = -3.0e38f; int bj = 0;
#pragma unroll
    for (int q = 0; q < COLS_; ++q) {
      int j = tid + q * TPB_SCAN;
      mind[q] = fminf(mind[q], row[j]);
      if (mind[q] > bv) { bv = mind[q]; bj = j; }
    }
    wave_amax(bv, bj);
    if (lane == 0) { s_wv[wave] = bv; s_wi[wave] = bj; }
    __syncthreads();
    if (wave == 0) {
      bv = s_wv[lane]; bj = s_wi[lane];
      wave_amax(bv, bj);
      if (lane == 0) s_last = bj;
    }
    __syncthreads();
    last = s_last;
  }
}

// ---------------------------------------------------------------------------
// Phase 3b (fallback if workspace can't hold the 268MB dist matrix):
// compute dist[last,:] on the fly as a GEMV from col-major features.
// ---------------------------------------------------------------------------
__global__ void ffps_scan_fused(const float* __restrict__ Fcm,
                                const float* __restrict__ sq,
                                int* __restrict__ out, int npoint) {
  int b = blockIdx.x, tid = threadIdx.x;
  int lane = tid & 31, wave = tid >> 5;
  __shared__ float fl[DP_];
  __shared__ float s_wv[32];
  __shared__ int   s_wi[32];
  __shared__ int   s_last;

  const float* fb  = Fcm + (size_t)b * DP_ * N_;
  const float* sqb = sq + (size_t)b * N_;
  float sql[COLS_], mind[COLS_];
#pragma unroll
  for (int q = 0; q < COLS_; ++q) {
    sql[q]  = sqb[tid + q * TPB_SCAN];
    mind[q] = 1e10f;
  }
  int last = 0;

  for (int s = 0; s < npoint; ++s) {
    if (tid < DP_) fl[tid] = fb[(size_t)tid * N_ + last];
    if (tid == 0) out[b * npoint + s] = last;
    __syncthreads();
    float sqlast = sqb[last];
    float dot[COLS_];
#pragma unroll
    for (int q = 0; q < COLS_; ++q) dot[q] = 0.f;
    for (int d = 0; d < DP_ - 1; ++d) {       // last pad col is zero
      float fv = fl[d];
      const float* colp = fb + (size_t)d * N_ + tid;
#pragma unroll
      for (int q = 0; q < COLS_; ++q) dot[q] += fv * colp[q * TPB_SCAN];
    }
    float bv = -3.0e38f; int bj = 0;
#pragma unroll
    for (int q = 0; q < COLS_; ++q) {
      float dval = sqlast + sql[q] - 2.0f * dot[q];
      mind[q] = fminf(mind[q], dval);
      int j = tid + q * TPB_SCAN;
      if (mind[q] > bv) { bv = mind[q]; bj = j; }
    }
    wave_amax(bv, bj);
    if (lane == 0) { s_wv[wave] = bv; s_wi[wave] = bj; }
    __syncthreads();
    if (wave == 0) {
      bv = s_wv[lane]; bj = s_wi[lane];
      wave_amax(bv, bj);
      if (lane == 0) s_last = bj;
    }
    __syncthreads();
    last = s_last;
  }
}

// ---------------------------------------------------------------------------
extern "C" void kernel_launch(void* const* d_in, const int* in_sizes, int n_in,
                              void* d_out, int out_size, void* d_ws, size_t ws_size,
                              hipStream_t stream) {
  const float* pts  = (const float*)d_in[0];   // [B, N, 3] f32
  const float* feat = (const float*)d_in[1];   // [B, C, N] f32
  int npoint = out_size / B_;                  // out: [B, npoint] int32

  const size_t frmBytes  = (size_t)B_ * N_ * DP_ * sizeof(float);   // ~8.65 MB
  const size_t sqBytes   = (size_t)B_ * N_ * sizeof(float);
  const size_t distBytes = (size_t)B_ * N_ * N_ * sizeof(float);    // 268 MB

  char* ws   = (char*)d_ws;
  float* Frm  = (float*)ws;
  float* sqv  = (float*)(ws + frmBytes);
  float* Fcm  = (float*)(ws + frmBytes + sqBytes);
  float* dist = (float*)(ws + frmBytes + sqBytes + frmBytes);
  bool full = ws_size >= 2 * frmBytes + sqBytes + distBytes;

  int* out = (int*)d_out;

  ffps_build<<<(B_ * N_ + 255) / 256, 256, 0, stream>>>(pts, feat, Frm, Fcm, sqv);

  if (full) {
    const int tiles = B_ * (N_ / 16) * (N_ / 16);   // 262144 waves
    ffps_gram_wmma<<<tiles / 4, 128, 0, stream>>>(Frm, sqv, dist);
    ffps_scan_pre<<<B_, TPB_SCAN, 0, stream>>>(dist, out, npoint);
  } else {
    ffps_scan_fused<<<B_, TPB_SCAN, 0, stream>>>(Fcm, sqv, out, npoint);
  }
}